// GCN_67396626809139
// MI455X (gfx1250) — compile-verified
//
#include <hip/hip_runtime.h>

typedef float v2f __attribute__((ext_vector_type(2)));
typedef float v8f __attribute__((ext_vector_type(8)));

template <int V> struct VecT;
template <> struct VecT<2> { typedef float type __attribute__((ext_vector_type(2))); };
template <> struct VecT<4> { typedef float type __attribute__((ext_vector_type(4))); };

static constexpr int NN   = 50000;    // nodes
static constexpr int NE   = 1600000;  // edges
static constexpr int CIN  = 384;
static constexpr int CHID = 128;
static constexpr int COUT = 64;
static constexpr int NSCAN_BLK = (NN + 255) / 256;   // 196

// ================= CSR build =================

__global__ __launch_bounds__(256) void k_zero_i32(int* p, int n) {
    int i = blockIdx.x * 256 + threadIdx.x;
    if (i < n) p[i] = 0;
}

__global__ __launch_bounds__(256) void k_count(const int* __restrict__ dst, int* cnt) {
    int e = blockIdx.x * 256 + threadIdx.x;
    if (e < NE) atomicAdd(&cnt[dst[e]], 1);
}

// Block-local exclusive scan of counts (Hillis-Steele in LDS) + per-block totals.
__global__ __launch_bounds__(256) void k_scan1(const int* __restrict__ cnt,
                                               int* __restrict__ offs,
                                               int* __restrict__ blksum) {
    __shared__ int buf[256];
    int t = threadIdx.x;
    int i = blockIdx.x * 256 + t;
    int v = (i < NN) ? cnt[i] : 0;
    buf[t] = v;
    __syncthreads();
#pragma unroll
    for (int off = 1; off < 256; off <<= 1) {
        int add = (t >= off) ? buf[t - off] : 0;
        __syncthreads();
        buf[t] += add;
        __syncthreads();
    }
    if (i < NN) offs[i] = buf[t] - v;              // exclusive
    if (t == 255) blksum[blockIdx.x] = buf[255];   // block total
}

// Single-block exclusive scan of the block sums.
__global__ __launch_bounds__(256) void k_scan2(int* __restrict__ blksum,
                                               int* __restrict__ blkoff) {
    __shared__ int buf[256];
    int t = threadIdx.x;
    int v = (t < NSCAN_BLK) ? blksum[t] : 0;
    buf[t] = v;
    __syncthreads();
#pragma unroll
    for (int off = 1; off < 256; off <<= 1) {
        int add = (t >= off) ? buf[t - off] : 0;
        __syncthreads();
        buf[t] += add;
        __syncthreads();
    }
    if (t < NSCAN_BLK) blkoff[t] = buf[t] - v;
}

// offs += block offset; init scatter cursor; dinv = rsqrt(deg) with self-loop.
__global__ __launch_bounds__(256) void k_scan3(int* __restrict__ offs,
                                               const int* __restrict__ blkoff,
                                               int* __restrict__ cursor,
                                               const int* __restrict__ cnt,
                                               float* __restrict__ dinv) {
    int i = blockIdx.x * 256 + threadIdx.x;
    if (i >= NN) return;
    int o = offs[i] + blkoff[blockIdx.x];
    offs[i]   = o;
    cursor[i] = o;
    dinv[i]   = rsqrtf((float)cnt[i] + 1.0f);      // +1 = self-loop
}

// Bucket each edge under its destination; pack (src, dinv[src]) into one int2.
__global__ __launch_bounds__(256) void k_scatter(const int* __restrict__ src,
                                                 const int* __restrict__ dst,
                                                 const float* __restrict__ dinv,
                                                 int* cursor,
                                                 int2* __restrict__ csr) {
    int e = blockIdx.x * 256 + threadIdx.x;
    if (e >= NE) return;
    int s = src[e];
    int d = dst[e];
    int pos = atomicAdd(&cursor[d], 1);
    int2 p;
    p.x = s;
    p.y = __float_as_int(dinv[s]);
    csr[pos] = p;                                   // single b64 store
}

// ================= dense GEMM via V_WMMA_F32_16X16X4_F32 =================
// One wave computes one 16x16 C tile; 8 waves per block.
template <int K, int NC, bool RELU>
__global__ __launch_bounds__(256) void k_gemm_wmma(const float* __restrict__ A,
                                                   const float* __restrict__ B,
                                                   float* __restrict__ C) {
    constexpr int NT = NC / 16;
    constexpr int MT = NN / 16;       // 3125, exact
    const int wave = threadIdx.x >> 5;
    const int lane = threadIdx.x & 31;
    const int tile = blockIdx.x * 8 + wave;
    if (tile >= MT * NT) return;
    const int mt = tile / NT;
    const int nt = tile - mt * NT;
    const int m  = lane & 15;
    const int hi = lane >> 4;

    const float* ap = A + (size_t)(mt * 16 + m) * K + 2 * hi;
    const float* bp = B + (size_t)(2 * hi) * NC + nt * 16 + m;

    v8f acc = {};
#pragma unroll 8
    for (int k = 0; k < K; k += 4) {
        v2f a = *(const v2f*)(ap + k);
        if (RELU) { a.x = fmaxf(a.x, 0.0f); a.y = fmaxf(a.y, 0.0f); }
        v2f b;
        b.x = bp[(size_t)k * NC];
        b.y = bp[(size_t)(k + 1) * NC];
        acc = __builtin_amdgcn_wmma_f32_16x16x4_f32(false, a, false, b,
                                                    (short)0, acc, false, false);
    }

    float* cp = C + (size_t)(mt * 16 + 8 * hi) * NC + nt * 16 + m;
#pragma unroll
    for (int r = 0; r < 8; ++r) cp[(size_t)r * NC] = acc[r];
}

// ================= CSR gather-reduce (no fp atomics, wide loads) =================
// One wave per node; lane owns VEC channels (b128/b64 row loads).
// out[i][c] = bias[c] + ( sum_j h[src_j][c]*dinv[src_j] + h[i][c]*dinv[i] ) * dinv[i]
template <int CH, int VEC>
__global__ __launch_bounds__(256) void k_gather(const int* __restrict__ offs,
                                                const int2* __restrict__ csr,
                                                const float* __restrict__ dinv,
                                                const float* __restrict__ h,
                                                const float* __restrict__ bias,
                                                float* __restrict__ out) {
    typedef typename VecT<VEC>::type vec;
    static_assert(VEC * 32 == CH, "one wave covers all channels");
    const int wave = threadIdx.x >> 5;
    const int lane = threadIdx.x & 31;
    const int i = blockIdx.x * 8 + wave;
    if (i >= NN) return;
    const int beg = offs[i];
    const int end = (i == NN - 1) ? NE : offs[i + 1];
    const size_t cbase = (size_t)lane * VEC;

    vec acc0 = {}, acc1 = {};
    int j = beg;
    for (; j + 1 < end; j += 2) {
        int2 p0 = csr[j];
        int2 p1 = csr[j + 1];
        float w0 = __int_as_float(p0.y);
        float w1 = __int_as_float(p1.y);
        vec r0 = *(const vec*)(h + (size_t)p0.x * CH + cbase);
        vec r1 = *(const vec*)(h + (size_t)p1.x * CH + cbase);
#pragma unroll
        for (int v = 0; v < VEC; ++v) {
            acc0[v] = fmaf(r0[v], w0, acc0[v]);
            acc1[v] = fmaf(r1[v], w1, acc1[v]);
        }
    }
    if (j < end) {
        int2 p0 = csr[j];
        float w0 = __int_as_float(p0.y);
        vec r0 = *(const vec*)(h + (size_t)p0.x * CH + cbase);
#pragma unroll
        for (int v = 0; v < VEC; ++v) acc0[v] = fmaf(r0[v], w0, acc0[v]);
    }

    float di = dinv[i];
    vec self = *(const vec*)(h + (size_t)i * CH + cbase);
    vec bv   = *(const vec*)(bias + cbase);
    vec o;
#pragma unroll
    for (int v = 0; v < VEC; ++v)
        o[v] = fmaf(fmaf(self[v], di, acc0[v] + acc1[v]), di, bv[v]);
    *(vec*)(out + (size_t)i * CH + cbase) = o;
}

// ================= fallback atomic path (small workspace) =================

__global__ __launch_bounds__(256) void k_deg_initf(float* deg) {
    int i = blockIdx.x * 256 + threadIdx.x;
    if (i < NN) deg[i] = 1.0f;
}
__global__ __launch_bounds__(256) void k_deg_countf(const int* __restrict__ dst, float* deg) {
    int e = blockIdx.x * 256 + threadIdx.x;
    if (e < NE)
        __hip_atomic_fetch_add(&deg[dst[e]], 1.0f, __ATOMIC_RELAXED, __HIP_MEMORY_SCOPE_AGENT);
}
__global__ __launch_bounds__(256) void k_dinvf(float* d) {
    int i = blockIdx.x * 256 + threadIdx.x;
    if (i < NN) d[i] = rsqrtf(d[i]);
}
template <int CH>
__global__ __launch_bounds__(256) void k_self_bias(const float* __restrict__ h,
                                                   const float* __restrict__ dinv,
                                                   const float* __restrict__ bias,
                                                   float* __restrict__ out) {
    int idx = blockIdx.x * 256 + threadIdx.x;
    if (idx >= NN * CH) return;
    int i = idx / CH;
    int c = idx & (CH - 1);
    float di = dinv[i];
    out[idx] = bias[c] + h[idx] * di * di;
}
template <int CH, int LOG2CH>
__global__ __launch_bounds__(256) void k_edge_agg(const int* __restrict__ src,
                                                  const int* __restrict__ dst,
                                                  const float* __restrict__ dinv,
                                                  const float* __restrict__ h,
                                                  float* out) {
    unsigned idx = blockIdx.x * 256 + threadIdx.x;
    unsigned e = idx >> LOG2CH;
    unsigned c = idx & (CH - 1);
    if (e >= NE) return;
    int s = src[e];
    int d = dst[e];
    float v = h[(size_t)s * CH + c] * dinv[s] * dinv[d];
    __hip_atomic_fetch_add(&out[(size_t)d * CH + c], v, __ATOMIC_RELAXED,
                           __HIP_MEMORY_SCOPE_AGENT);
}

// ================= launcher =================

extern "C" void kernel_launch(void* const* d_in, const int* in_sizes, int n_in,
                              void* d_out, int out_size, void* d_ws, size_t ws_size,
                              hipStream_t stream) {
    const float* x  = (const float*)d_in[0];
    const int*   ei = (const int*)d_in[1];
    const float* W1 = (const float*)d_in[2];
    const float* b1 = (const float*)d_in[3];
    const float* W2 = (const float*)d_in[4];
    const float* b2 = (const float*)d_in[5];
    const int* src = ei;
    const int* dst = ei + NE;
    float* out = (float*)d_out;

    const size_t fH1 = (size_t)NN * CHID, fAG = (size_t)NN * CHID, fH2 = (size_t)NN * COUT;

    // CSR-path workspace (bytes)
    const size_t csr_bytes =
        sizeof(int) * (3 * (size_t)NN + 2 * 256) +
        sizeof(int2) * (size_t)NE +
        sizeof(float) * ((size_t)NN + fH1 + fAG + fH2);

    if (ws_size >= csr_bytes) {
        // ---- layout (int block is 602048 B -> int2 stays 8B-aligned) ----
        int* cnt    = (int*)d_ws;
        int* offs   = cnt + NN;
        int* cursor = offs + NN;
        int* blksum = cursor + NN;
        int* blkoff = blksum + 256;
        int2* csr   = (int2*)(blkoff + 256);
        float* dinv = (float*)(csr + NE);
        float* h1   = dinv + NN;
        float* agg1 = h1 + fH1;
        float* h2   = agg1 + fAG;

        // ---- CSR build ----
        k_zero_i32<<<(NN + 255) / 256, 256, 0, stream>>>(cnt, NN);
        k_count<<<(NE + 255) / 256, 256, 0, stream>>>(dst, cnt);
        k_scan1<<<NSCAN_BLK, 256, 0, stream>>>(cnt, offs, blksum);
        k_scan2<<<1, 256, 0, stream>>>(blksum, blkoff);
        k_scan3<<<NSCAN_BLK, 256, 0, stream>>>(offs, blkoff, cursor, cnt, dinv);
        k_scatter<<<(NE + 255) / 256, 256, 0, stream>>>(src, dst, dinv, cursor, csr);

        // ---- layer 1 ----
        int tiles1 = (NN / 16) * (CHID / 16);
        k_gemm_wmma<CIN, CHID, false><<<(tiles1 + 7) / 8, 256, 0, stream>>>(x, W1, h1);
        k_gather<CHID, 4><<<(NN + 7) / 8, 256, 0, stream>>>(offs, csr, dinv, h1, b1, agg1);

        // ---- layer 2 ----
        int tiles2 = (NN / 16) * (COUT / 16);
        k_gemm_wmma<CHID, COUT, true><<<(tiles2 + 7) / 8, 256, 0, stream>>>(agg1, W2, h2);
        k_gather<COUT, 2><<<(NN + 7) / 8, 256, 0, stream>>>(offs, csr, dinv, h2, b2, out);
    } else {
        // ---- fallback: edge-parallel atomics ----
        float* dinv = (float*)d_ws;
        float* h1   = dinv + NN;
        float* agg1 = h1 + fH1;
        float* h2   = agg1 + fAG;

        k_deg_initf<<<(NN + 255) / 256, 256, 0, stream>>>(dinv);
        k_deg_countf<<<(NE + 255) / 256, 256, 0, stream>>>(dst, dinv);
        k_dinvf<<<(NN + 255) / 256, 256, 0, stream>>>(dinv);

        int tiles1 = (NN / 16) * (CHID / 16);
        k_gemm_wmma<CIN, CHID, false><<<(tiles1 + 7) / 8, 256, 0, stream>>>(x, W1, h1);
        k_self_bias<CHID><<<(NN * CHID + 255) / 256, 256, 0, stream>>>(h1, dinv, b1, agg1);
        k_edge_agg<CHID, 7><<<(int)(((size_t)NE * CHID) / 256), 256, 0, stream>>>(src, dst, dinv, h1, agg1);

        int tiles2 = (NN / 16) * (COUT / 16);
        k_gemm_wmma<CHID, COUT, true><<<(tiles2 + 7) / 8, 256, 0, stream>>>(agg1, W2, h2);
        k_self_bias<COUT><<<(NN * COUT + 255) / 256, 256, 0, stream>>>(h2, dinv, b2, out);
        k_edge_agg<COUT, 6><<<(int)(((size_t)NE * COUT) / 256), 256, 0, stream>>>(src, dst, dinv, h2, out);
    }
}